// MultiBoxLoss_89781996355747
// MI455X (gfx1250) — compile-verified
//
#include <hip/hip_runtime.h>
#include <stdint.h>

// Problem constants (match reference).
#define NB 128     // batch
#define NP 8732    // priors
#define NM 16      // objects per image
#define NC 21      // classes
#define NEGR 3     // hard-negative ratio
#define BS 1024    // threads per block (32 wave32 waves)
#define TILE 240   // priors per staged score tile (TILE*NC*4 = 20160 B)
#define NWAVES (BS / 32)

#if defined(__has_builtin)
#if __has_builtin(__builtin_amdgcn_tensor_load_to_lds) && \
    __has_builtin(__builtin_amdgcn_s_wait_tensorcnt)
#define USE_TDM 1
#endif
#if __has_builtin(__builtin_amdgcn_global_load_async_to_lds_b128) && \
    __has_builtin(__builtin_amdgcn_s_wait_asynccnt)
#define USE_ASYNC_LDS 1
#endif
#endif

typedef int v4i_t __attribute__((ext_vector_type(4)));
typedef __attribute__((address_space(1))) v4i_t gv4i_t;
typedef __attribute__((address_space(3))) v4i_t lv4i_t;
#ifdef USE_TDM
typedef unsigned int u32x4_t __attribute__((ext_vector_type(4)));
typedef int i32x4_t __attribute__((ext_vector_type(4)));
typedef int i32x8_t __attribute__((ext_vector_type(8)));
typedef __attribute__((address_space(3))) float lfloat_t;
#endif

__device__ __forceinline__ float wredf(float v) {
  #pragma unroll
  for (int o = 16; o > 0; o >>= 1) v += __shfl_xor(v, o, 32);
  return v;
}
__device__ __forceinline__ unsigned wredu(unsigned v) {
  #pragma unroll
  for (int o = 16; o > 0; o >>= 1) v += __shfl_xor(v, o, 32);
  return v;
}

// Block-wide reductions (uniform control flow; scratch has NWAVES slots).
__device__ __forceinline__ float bredf(float v, float* sc) {
  v = wredf(v);
  const int lane = threadIdx.x & 31, w = threadIdx.x >> 5;
  if (lane == 0) sc[w] = v;
  __syncthreads();
  if (threadIdx.x < 32) {
    float r = sc[threadIdx.x];
    r = wredf(r);
    if (threadIdx.x == 0) sc[0] = r;
  }
  __syncthreads();
  const float r = sc[0];
  __syncthreads();
  return r;
}
__device__ __forceinline__ unsigned bredu(unsigned v, unsigned* sc) {
  v = wredu(v);
  const int lane = threadIdx.x & 31, w = threadIdx.x >> 5;
  if (lane == 0) sc[w] = v;
  __syncthreads();
  if (threadIdx.x < 32) {
    unsigned r = sc[threadIdx.x];
    r = wredu(r);
    if (threadIdx.x == 0) sc[0] = r;
  }
  __syncthreads();
  const unsigned r = sc[0];
  __syncthreads();
  return r;
}

__global__ __launch_bounds__(BS) void mbox_main(
    const float* __restrict__ pred_locs,    // [B,P,4]
    const float* __restrict__ pred_scores,  // [B,P,C]
    const float* __restrict__ bboxes,       // [B,M,4] corner
    const int* __restrict__ labels,         // [B,M]
    const float* __restrict__ priors,       // [P,4] cxcy
    float* __restrict__ locPart,            // [B]
    float* __restrict__ confPart,           // [B]
    int* __restrict__ posPart)              // [B]
{
  __shared__ float4 s_box[NM];                       // corner boxes
  __shared__ int s_lab[NM];
  __shared__ unsigned long long s_best[NM];          // per-object best (iou_bits<<32)|~p
  __shared__ unsigned char s_code[NP];               // bit7 = positive, bits[4:0] = obj
  __shared__ float s_conf[NP];                       // negative CE values (0 at positives)
  __shared__ __align__(16) float s_scores[TILE * NC];
  __shared__ float s_scf[NWAVES];
  __shared__ unsigned s_scu[NWAVES];

  const int b = blockIdx.x;
  const int tid = threadIdx.x;

  if (tid < NM) {
    const float4 bx = reinterpret_cast<const float4*>(bboxes)[b * NM + tid];
    s_box[tid] = bx;
    s_lab[tid] = labels[b * NM + tid];
    s_best[tid] = 0ull;
  }
  __syncthreads();

  // ---- Phase 1: IoU matching ----
  unsigned long long bKey[NM];
  #pragma unroll
  for (int m = 0; m < NM; ++m) bKey[m] = 0ull;

  for (int p = tid; p < NP; p += BS) {
    const float4 pc = reinterpret_cast<const float4*>(priors)[p];
    const float px0 = pc.x - 0.5f * pc.z, py0 = pc.y - 0.5f * pc.w;
    const float px1 = pc.x + 0.5f * pc.z, py1 = pc.y + 0.5f * pc.w;
    const float pa = pc.z * pc.w;
    float biou = -1.0f;
    int bm = 0;
    #pragma unroll
    for (int m = 0; m < NM; ++m) {
      const float4 bx = s_box[m];
      const float ix0 = fmaxf(px0, bx.x), iy0 = fmaxf(py0, bx.y);
      const float ix1 = fminf(px1, bx.z), iy1 = fminf(py1, bx.w);
      const float iw = fmaxf(ix1 - ix0, 0.0f), ih = fmaxf(iy1 - iy0, 0.0f);
      const float inter = iw * ih;
      const float ba = (bx.z - bx.x) * (bx.w - bx.y);
      const float iou = inter / (pa + ba - inter);
      if (iou > biou) { biou = iou; bm = m; }  // strict > => first-occurrence argmax
      const unsigned long long key =
          ((unsigned long long)__float_as_uint(iou) << 32) |
          (unsigned long long)(0xFFFFFFFFu - (unsigned)p);  // ties -> lowest prior
      if (key > bKey[m]) bKey[m] = key;
    }
    s_code[p] = (unsigned char)((biou >= 0.5f ? 0x80 : 0) | bm);
  }
  // Reduce per-object best across wave, then LDS ds_max_u64 across waves.
  #pragma unroll
  for (int m = 0; m < NM; ++m) {
    unsigned long long k = bKey[m];
    #pragma unroll
    for (int o = 16; o > 0; o >>= 1) {
      const unsigned long long t = __shfl_xor(k, o, 32);
      if (t > k) k = t;
    }
    if ((tid & 31) == 0) atomicMax(&s_best[m], k);
  }
  __syncthreads();

  // ---- Phase 2: forced matches (obj m claims its best prior, iou := 1) ----
  if (tid < NM) {
    const unsigned p = 0xFFFFFFFFu - (unsigned)(s_best[tid] & 0xFFFFFFFFull);
    s_code[p] = (unsigned char)(0x80 | tid);
  }
  __syncthreads();

  // ---- Phase 3: cross-entropy + SmoothL1; stream scores via TDM / async->LDS ----
  float locSum = 0.0f, confPos = 0.0f;
  unsigned posCnt = 0;
  const size_t scoreBase = (size_t)b * NP * NC;  // floats

#ifdef USE_TDM
  // LDS byte address of the staging buffer (AS(3) pointer offset).
  const unsigned ldsAddr = (unsigned)(unsigned long long)(lfloat_t*)&s_scores[0];
#endif

  for (int t0 = 0; t0 < NP; t0 += TILE) {
    const int rows = min(TILE, NP - t0);
#ifdef USE_TDM
    if (tid == 0) {  // one wave issues the DMA; TDM ignores EXEC
      const unsigned n = (unsigned)(rows * NC);  // elements (4B each), <= 5040
      const unsigned long long ga =
          (unsigned long long)(const void*)(pred_scores + scoreBase + (size_t)t0 * NC);
      // D# group0: count=1 | lds_addr | global_addr(57b) | type=2
      const u32x4_t g0 = {1u, ldsAddr, (unsigned)(ga & 0xFFFFFFFFull),
                          (unsigned)((ga >> 32) & 0x01FFFFFFull) | (2u << 30)};
      // D# group1: data_size=4B; tensor_dim0=n; tensor_dim1=1; tile_dim0=n;
      //            tile_dim1=1; tensor_dim0_stride=n (single line)
      const i32x8_t g1 = {(int)(2u << 16),          // wgmask=0, data_size=2(4B)
                          (int)((n & 0xFFFFu) << 16),  // tdim0[15:0] @ bits63:48
                          (int)((n >> 16) | (1u << 16)),  // tdim0[31:16], tdim1=1
                          (int)(n << 16),           // tdim1 hi=0, tile_dim0=n
                          1,                        // tile_dim1=1, tile_dim2=0
                          (int)n,                   // stride0[31:0]
                          0, 0};                    // stride0 hi, stride1
      const i32x4_t gz = {0, 0, 0, 0};              // groups 2/3 unused (<=2D)
      const i32x8_t gz8 = {0, 0, 0, 0, 0, 0, 0, 0}; // extra group (6-arg form)
      __builtin_amdgcn_tensor_load_to_lds(g0, g1, gz, gz, gz8, 0);
      __builtin_amdgcn_s_wait_tensorcnt(0);
    }
#else
    const int chunks = rows * NC / 4;  // 16-byte chunks (rows multiple of 4)
#ifdef USE_ASYNC_LDS
    for (int c = tid; c < chunks; c += BS) {
      const float* g = pred_scores + scoreBase + (size_t)t0 * NC + (size_t)c * 4;
      __builtin_amdgcn_global_load_async_to_lds_b128(
          (gv4i_t*)g, (lv4i_t*)&s_scores[c * 4], 0, 0);
    }
    __builtin_amdgcn_s_wait_asynccnt(0);
#else
    for (int c = tid; c < chunks; c += BS) {
      const float4 v = *reinterpret_cast<const float4*>(
          pred_scores + scoreBase + (size_t)t0 * NC + (size_t)c * 4);
      *reinterpret_cast<float4*>(&s_scores[c * 4]) = v;
    }
#endif
#endif
    __syncthreads();

    if (tid < rows) {
      const int p = t0 + tid;
      const float* row = &s_scores[tid * NC];
      float mx = row[0];
      #pragma unroll
      for (int k = 1; k < NC; ++k) mx = fmaxf(mx, row[k]);
      float se = 0.0f;
      #pragma unroll
      for (int k = 0; k < NC; ++k) se += __expf(row[k] - mx);
      const float lse = mx + __logf(se);

      const unsigned code = s_code[p];
      const int obj = code & 0x1F;
      const bool pos = (code & 0x80) != 0;
      const int tc = pos ? s_lab[obj] : 0;
      float conf = lse - row[tc];
      conf = fmaxf(conf, 0.0f);

      if (pos) {
        posCnt++;
        confPos += conf;
        s_conf[p] = 0.0f;
        // Reference feeds CORNER boxes into cxcy_to_gcxgcy; replicate exactly.
        const float4 pc = reinterpret_cast<const float4*>(priors)[p];
        const float4 pl = reinterpret_cast<const float4*>(pred_locs)[(size_t)b * NP + p];
        const float4 bx = s_box[obj];
        const float g0 = (bx.x - pc.x) * 10.0f / pc.z;
        const float g1 = (bx.y - pc.y) * 10.0f / pc.w;
        const float g2 = 5.0f * __logf(bx.z / pc.z);
        const float g3 = 5.0f * __logf(bx.w / pc.w);
        float d;
        d = fabsf(pl.x - g0); locSum += (d < 1.0f) ? 0.5f * d * d : d - 0.5f;
        d = fabsf(pl.y - g1); locSum += (d < 1.0f) ? 0.5f * d * d : d - 0.5f;
        d = fabsf(pl.z - g2); locSum += (d < 1.0f) ? 0.5f * d * d : d - 0.5f;
        d = fabsf(pl.w - g3); locSum += (d < 1.0f) ? 0.5f * d * d : d - 0.5f;
      } else {
        s_conf[p] = conf;
      }
    }
    __syncthreads();  // protect s_scores before next tile's DMA/async writes
  }

  const unsigned numPos = bredu(posCnt, s_scu);
  const float cPos = bredf(confPos, s_scf);
  const float lSum = bredf(locSum, s_scf);

  // ---- Phase 4: exact top-K sum of negatives via bit-pattern bisection ----
  // Values are >= 0 so uint bit order == numeric order.
  const unsigned K = NEGR * numPos;
  unsigned lo = 0, hi = 0x7F800000u;  // [0, +inf)
  while (lo < hi) {  // lo/hi uniform across block -> barriers inside are safe
    const unsigned mid = (lo + hi) >> 1;
    unsigned cnt = 0;
    for (int p = tid; p < NP; p += BS)
      cnt += (__float_as_uint(s_conf[p]) > mid) ? 1u : 0u;
    cnt = bredu(cnt, s_scu);
    if (cnt <= K) hi = mid; else lo = mid + 1;
  }
  unsigned cntGt = 0;
  float sumGt = 0.0f;
  for (int p = tid; p < NP; p += BS) {
    const float v = s_conf[p];
    if (__float_as_uint(v) > lo) { cntGt++; sumGt += v; }
  }
  cntGt = bredu(cntGt, s_scu);
  sumGt = bredf(sumGt, s_scf);
  const float negSum = sumGt + (float)(K - cntGt) * __uint_as_float(lo);

  if (tid == 0) {
    locPart[b] = lSum;
    confPart[b] = cPos + negSum;
    posPart[b] = (int)numPos;
  }
}

__global__ __launch_bounds__(128) void mbox_final(
    const float* __restrict__ locPart, const float* __restrict__ confPart,
    const int* __restrict__ posPart, float* __restrict__ out)
{
  __shared__ float s[12];
  const int tid = threadIdx.x;
  float l = locPart[tid];
  float c = confPart[tid];
  float n = (float)posPart[tid];
  l = wredf(l);
  c = wredf(c);
  n = wredf(n);
  const int w = tid >> 5;
  if ((tid & 31) == 0) { s[w] = l; s[4 + w] = c; s[8 + w] = n; }
  __syncthreads();
  if (tid == 0) {
    const float L = s[0] + s[1] + s[2] + s[3];
    const float C = s[4] + s[5] + s[6] + s[7];
    const float N = s[8] + s[9] + s[10] + s[11];
    out[0] = C / N + L / (4.0f * N);
  }
}

extern "C" void kernel_launch(void* const* d_in, const int* in_sizes, int n_in,
                              void* d_out, int out_size, void* d_ws, size_t ws_size,
                              hipStream_t stream) {
  (void)in_sizes; (void)n_in; (void)out_size; (void)ws_size;
  const float* pred_locs   = (const float*)d_in[0];
  const float* pred_scores = (const float*)d_in[1];
  const float* bboxes      = (const float*)d_in[2];
  const int*   labels      = (const int*)d_in[3];
  const float* priors      = (const float*)d_in[4];

  float* locPart  = (float*)d_ws;
  float* confPart = locPart + NB;
  int*   posPart  = (int*)(confPart + NB);

  mbox_main<<<NB, BS, 0, stream>>>(pred_locs, pred_scores, bboxes, labels,
                                   priors, locPart, confPart, posPart);
  mbox_final<<<1, NB, 0, stream>>>(locPart, confPart, posPart, (float*)d_out);
}